// ComaAtt_34918084116766
// MI455X (gfx1250) — compile-verified
//
#include <hip/hip_runtime.h>
#include <hip/hip_bf16.h>
#include <cstdint>
#include <cstddef>

// ---------------------------------------------------------------------------
// CoMA mesh autoencoder forward for MI455X (gfx1250, wave32).
// Dense per-node feature transforms: V_WMMA_F32_16X16X4_F32, templated on
// (FIN, FOUT, MODE) so K-loops fully unroll, loads are unconditional, and the
// writeback is straight-line (no runtime-flag branches).
// Sparse propagate/pool: global_atomic_add_f32 / gathers.
// ---------------------------------------------------------------------------

typedef __attribute__((ext_vector_type(2))) float v2f;
typedef __attribute__((ext_vector_type(8))) float v8f;

#define CDIV(a, b) (((a) + (b) - 1) / (b))

// ------------------------------ utility kernels ----------------------------

__global__ void k_zero(float* __restrict__ p, long long n) {
  long long i = (long long)blockIdx.x * blockDim.x + threadIdx.x;
  if (i < n) p[i] = 0.0f;
}

__global__ void k_negcopy(float* __restrict__ dst, const float* __restrict__ src,
                          long long n) {
  long long i = (long long)blockIdx.x * blockDim.x + threadIdx.x;
  if (i < n) dst[i] = -src[i];
}

__global__ void k_deg_count(const int* __restrict__ ei, float* __restrict__ deg, int E) {
  int e = blockIdx.x * blockDim.x + threadIdx.x;
  if (e < E) atomicAdd(&deg[ei[e]], 1.0f);
}

__global__ void k_deg_inv(float* __restrict__ deg, int N) {
  int i = blockIdx.x * blockDim.x + threadIdx.x;
  if (i < N) {
    float d = deg[i];
    deg[i] = (d > 0.0f) ? (1.0f / sqrtf(d)) : 0.0f;
  }
}

__global__ void k_edge_norm(const int* __restrict__ ei, const float* __restrict__ dis,
                            float* __restrict__ nrm, int E) {
  int e = blockIdx.x * blockDim.x + threadIdx.x;
  if (e < E) nrm[e] = dis[ei[e]] * dis[ei[E + e]];
}

// out[b, row[e], f] += scale * norm[e] * x[b, col[e], f]   (B = 4 fixed)
__global__ void k_propagate(const int* __restrict__ ei, const float* __restrict__ nrm,
                            const float* __restrict__ x, float* __restrict__ out,
                            int E, int N, int F, float scale) {
  long long tid = (long long)blockIdx.x * blockDim.x + threadIdx.x;
  long long total = (long long)E * F;
  if (tid >= total) return;
  int e = (int)(tid / F);
  int f = (int)(tid % F);
  int r = ei[e];
  int c = ei[E + e];
  float w = scale * nrm[e];
  size_t nf = (size_t)N * F;
#pragma unroll
  for (int b = 0; b < 4; ++b) {
    float v = w * x[(size_t)b * nf + (size_t)c * F + f];
    atomicAdd(&out[(size_t)b * nf + (size_t)r * F + f], v);
  }
}

// out[b,n,f] = sum_k w[n,k] * x[b, idx[n,k], f]   (3 taps)
__global__ void k_pool(const float* __restrict__ x, const int* __restrict__ idx,
                       const float* __restrict__ w, float* __restrict__ out,
                       int Nout, int Nin, int F) {
  long long tid = (long long)blockIdx.x * blockDim.x + threadIdx.x;
  long long total = 4LL * Nout * F;
  if (tid >= total) return;
  int f = (int)(tid % F);
  int n = (int)((tid / F) % Nout);
  int b = (int)(tid / ((long long)F * Nout));
  size_t nfin = (size_t)Nin * F;
  float s = 0.0f;
#pragma unroll
  for (int k = 0; k < 3; ++k) {
    int j = idx[n * 3 + k];
    s += w[n * 3 + k] * x[(size_t)b * nfin + (size_t)j * F + f];
  }
  out[((size_t)b * Nout + n) * F + f] = s;
}

// z[b,j] = relu(sum_i h[b,i] * w[i,j] + bias[j]); h:[4,8192], w:[8192,64]
__global__ void k_dense_enc(const float* __restrict__ h, const float* __restrict__ w,
                            const float* __restrict__ bias, float* __restrict__ z) {
  int b = blockIdx.x >> 6;
  int j = blockIdx.x & 63;
  float s = 0.0f;
  for (int i = threadIdx.x; i < 8192; i += blockDim.x)
    s += h[b * 8192 + i] * w[(size_t)i * 64 + j];
  __shared__ float red[256];
  red[threadIdx.x] = s;
  __syncthreads();
  for (int off = 128; off > 0; off >>= 1) {
    if (threadIdx.x < off) red[threadIdx.x] += red[threadIdx.x + off];
    __syncthreads();
  }
  if (threadIdx.x == 0) {
    float v = red[0] + bias[j];
    z[b * 64 + j] = v > 0.0f ? v : 0.0f;
  }
}

// h[b,j] = relu(sum_i z[b,i] * w[i,j] + bias[j]); z:[4,64], w:[64,8192]
__global__ void k_dense_dec(const float* __restrict__ z, const float* __restrict__ w,
                            const float* __restrict__ bias, float* __restrict__ h) {
  long long tid = (long long)blockIdx.x * blockDim.x + threadIdx.x;
  if (tid >= 4LL * 8192) return;
  int b = (int)(tid >> 13);
  int j = (int)(tid & 8191);
  float s = bias[j];
  for (int i = 0; i < 64; ++i) s += z[b * 64 + i] * w[(size_t)i * 8192 + j];
  h[tid] = s > 0.0f ? s : 0.0f;
}

// ------------------------ WMMA feature-transform GEMM ----------------------

// Always-in-bounds load + select: no EXEC-masked branch, just v_cndmask.
__device__ __forceinline__ float ldsel(const float* __restrict__ p, int idx, bool ok) {
  float v = p[ok ? idx : 0];
  return ok ? v : 0.0f;
}

// out[M, FOUT] (op)= T[M, FIN] @ W[FIN, FOUT]
// MODE: 0 = out = bias + T@W ; 1 = out = T@W ; 2 = out += T@W ;
//       3 = out = relu(out + T@W)
// Each wave owns TM=4 consecutive 16-row tiles x one 16-col tile; the B
// (weight) fragments are loaded once and reused across the 4 row tiles.
// Fragment layouts per CDNA5 ISA:
//   A 16x4: lane l -> row l%16, K = {0,1} (lanes 0-15) / {2,3} (lanes 16-31)
//   B 4x16: mirrored K striping, N = lane%16
//   C/D 16x16: VGPR r -> M = r + 8*(lane>=16), N = lane%16
template <int FIN, int FOUT, int MODE>
__global__ __launch_bounds__(256) void k_cheb_gemm(
    const float* __restrict__ T, const float* __restrict__ W,
    const float* __restrict__ bias, float* __restrict__ out, int M) {
  constexpr int KC = (FIN + 3) / 4;      // K chunks of 4
  constexpr int NTN = (FOUT + 15) / 16;  // 16-wide column tiles
  constexpr int TM = 4;                  // row tiles per wave

  int wave = (int)((blockIdx.x * blockDim.x + threadIdx.x) >> 5);
  int lane = threadIdx.x & 31;
  int groups = (M >> 4) / TM;            // M/16 is always a multiple of 4 here
  if (wave >= groups * NTN) return;      // wave-uniform exit

  int tn = wave % NTN;
  int gm = wave / NTN;
  int m0 = gm * (16 * TM);
  int n0 = tn * 16;
  int lm = lane & 15;
  int khalf = (lane >> 4) << 1;  // 0 for lanes 0-15, 2 for lanes 16-31
  int mhi = (lane >> 4) << 3;    // C/D: M = r (+8 for upper half-wave)
  int ncol = n0 + lm;
  bool colok = (FOUT % 16 == 0) || (ncol < FOUT);

  // Load B (weight) fragments once; bias value once (MODE 0 only).
  v2f bf[KC];
#pragma unroll
  for (int kc = 0; kc < KC; ++kc) {
    int ka = kc * 4 + khalf;
    if constexpr (FIN % 4 == 0 && FOUT % 16 == 0) {
      bf[kc].x = W[ka * FOUT + ncol];
      bf[kc].y = W[(ka + 1) * FOUT + ncol];
    } else {
      bf[kc].x = ldsel(W, ka * FOUT + ncol, colok && (ka < FIN));
      bf[kc].y = ldsel(W, (ka + 1) * FOUT + ncol, colok && (ka + 1 < FIN));
    }
  }
  float bv = 0.0f;
  if constexpr (MODE == 0) bv = ldsel(bias, ncol, colok);

#pragma unroll
  for (int t = 0; t < TM; ++t) {
    int mb = m0 + t * 16;
    const float* trow = T + (size_t)(mb + lm) * FIN;
    v8f c = {0.f, 0.f, 0.f, 0.f, 0.f, 0.f, 0.f, 0.f};
#pragma unroll
    for (int kc = 0; kc < KC; ++kc) {
      int ka = kc * 4 + khalf;
      v2f a;
      if constexpr (FIN % 4 == 0) {
        a = *(const v2f*)(trow + ka);  // 8B-aligned: even row stride, even ka
      } else {
        a.x = ldsel(trow, ka, ka < FIN);
        a.y = ldsel(trow, ka + 1, ka + 1 < FIN);
      }
      c = __builtin_amdgcn_wmma_f32_16x16x4_f32(false, a, false, bf[kc],
                                                (short)0, c, false, false);
    }
    float* orow = out + (size_t)(mb + mhi) * FOUT + ncol;
    if (colok) {
#pragma unroll
      for (int r = 0; r < 8; ++r) {
        float v = c[r];
        if constexpr (MODE == 0) v += bv;
        if constexpr (MODE == 2 || MODE == 3) v += orow[(size_t)r * FOUT];
        if constexpr (MODE == 3) v = v > 0.0f ? v : 0.0f;
        orow[(size_t)r * FOUT] = v;
      }
    }
  }
}

// ------------------------------ host helpers -------------------------------

template <int FIN, int FOUT>
static void launch_gemm_sh(const float* T, const float* W, const float* bias,
                           float* out, int M, int mode, hipStream_t s) {
  constexpr int TM = 4;
  constexpr int NTN = (FOUT + 15) / 16;
  long long waves = (long long)((M / 16) / TM) * NTN;
  unsigned blocks = (unsigned)CDIV(waves * 32, 256);
  switch (mode) {
    case 0: k_cheb_gemm<FIN, FOUT, 0><<<blocks, 256, 0, s>>>(T, W, bias, out, M); break;
    case 1: k_cheb_gemm<FIN, FOUT, 1><<<blocks, 256, 0, s>>>(T, W, bias, out, M); break;
    case 2: k_cheb_gemm<FIN, FOUT, 2><<<blocks, 256, 0, s>>>(T, W, bias, out, M); break;
    default: k_cheb_gemm<FIN, FOUT, 3><<<blocks, 256, 0, s>>>(T, W, bias, out, M); break;
  }
}

static void launch_gemm(const float* T, const float* W, const float* bias,
                        float* out, int M, int Fin, int Fout, int mode,
                        hipStream_t s) {
  if (Fin == 3 && Fout == 16)       launch_gemm_sh<3, 16>(T, W, bias, out, M, mode, s);
  else if (Fin == 16 && Fout == 16) launch_gemm_sh<16, 16>(T, W, bias, out, M, mode, s);
  else if (Fin == 16 && Fout == 32) launch_gemm_sh<16, 32>(T, W, bias, out, M, mode, s);
  else if (Fin == 32 && Fout == 16) launch_gemm_sh<32, 16>(T, W, bias, out, M, mode, s);
  else if (Fin == 16 && Fout == 3)  launch_gemm_sh<16, 3>(T, W, bias, out, M, mode, s);
}

// out = relu?( sum_k T_k(x) @ W[k] + bias )  — Chebyshev conv at one level
static void run_conv(const float* x, const int* ei, const float* nrm,
                     const float* W /*[6,Fin,Fout]*/, const float* bias,
                     float* out, float* b0, float* b1, float* b2,
                     int N, int Fin, int Fout, int relu, hipStream_t s) {
  int E = 16 * N;
  int M = 4 * N;
  long long tsz = (long long)M * Fin;
  size_t wstride = (size_t)Fin * Fout;

  // k = 0: out = bias + x @ W0   (or out = x @ W0 when no bias)
  launch_gemm(x, W, bias, out, M, Fin, Fout, bias ? 0 : 1, s);

  // T1 = L x
  k_zero<<<(unsigned)CDIV(tsz, 256), 256, 0, s>>>(b0, tsz);
  k_propagate<<<(unsigned)CDIV((long long)E * Fin, 256), 256, 0, s>>>(
      ei, nrm, x, b0, E, N, Fin, 1.0f);
  launch_gemm(b0, W + wstride, nullptr, out, M, Fin, Fout, 2, s);

  // T_k = 2 L T_{k-1} - T_{k-2}
  const float* p0 = x;
  float* prev = b0;
  float* bufs[3] = {b0, b1, b2};
  for (int k = 2; k < 6; ++k) {
    float* tk = bufs[(k - 1) % 3];  // k=2->b1, 3->b2, 4->b0, 5->b1
    k_negcopy<<<(unsigned)CDIV(tsz, 256), 256, 0, s>>>(tk, p0, tsz);
    k_propagate<<<(unsigned)CDIV((long long)E * Fin, 256), 256, 0, s>>>(
        ei, nrm, prev, tk, E, N, Fin, 2.0f);
    int last = (k == 5);
    launch_gemm(tk, W + (size_t)k * wstride, nullptr, out, M, Fin, Fout,
                (last && relu) ? 3 : 2, s);
    p0 = prev;
    prev = tk;
  }
}

// ------------------------------ entry point --------------------------------

extern "C" void kernel_launch(void* const* d_in, const int* in_sizes, int n_in,
                              void* d_out, int out_size, void* d_ws, size_t ws_size,
                              hipStream_t stream) {
  (void)in_sizes; (void)n_in; (void)out_size; (void)ws_size;
  static const int NSs[5] = {65536, 16384, 4096, 1024, 256};
  static const int FE[5] = {3, 16, 16, 16, 32};
  static const int FD[6] = {32, 16, 16, 16, 16, 3};

  const float* x = (const float*)d_in[0];
  const int* ei[5];
  for (int l = 0; l < 5; ++l) ei[l] = (const int*)d_in[1 + l];
  const int* down_i[4]; const float* down_w[4];
  const int* up_i[4];   const float* up_w[4];
  for (int i = 0; i < 4; ++i) {
    down_i[i] = (const int*)d_in[6 + 4 * i];
    down_w[i] = (const float*)d_in[7 + 4 * i];
    up_i[i]   = (const int*)d_in[8 + 4 * i];
    up_w[i]   = (const float*)d_in[9 + 4 * i];
  }
  const float* encW[4]; const float* encB[4];
  for (int i = 0; i < 4; ++i) {
    encW[i] = (const float*)d_in[22 + 2 * i];
    encB[i] = (const float*)d_in[23 + 2 * i];
  }
  const float* decW[5];
  for (int i = 0; i < 5; ++i) decW[i] = (const float*)d_in[30 + i];
  const float* decB[4];
  for (int i = 0; i < 4; ++i) decB[i] = (const float*)d_in[35 + i];
  const float* enc_w = (const float*)d_in[39];
  const float* enc_b = (const float*)d_in[40];
  const float* dec_w = (const float*)d_in[41];
  const float* dec_b = (const float*)d_in[42];

  // workspace carving (floats)
  float* ws = (float*)d_ws;
  size_t off = 0;
  float* nrm[5];
  for (int l = 0; l < 5; ++l) { nrm[l] = ws + off; off += (size_t)16 * NSs[l]; }
  float* deg = ws + off; off += 65536;
  const size_t ACT = (size_t)4 * 65536 * 16;
  float* actA = ws + off; off += ACT;
  float* actB = ws + off; off += ACT;
  float* tb0 = ws + off; off += ACT;
  float* tb1 = ws + off; off += ACT;
  float* tb2 = ws + off; off += ACT;
  float* zbuf = ws + off; off += 256;

  // ---- per-level edge norms: D^-1/2 A D^-1/2 ----
  for (int l = 0; l < 5; ++l) {
    int N = NSs[l], E = 16 * N;
    k_zero<<<(unsigned)CDIV((long long)N, 256), 256, 0, stream>>>(deg, N);
    k_deg_count<<<(unsigned)CDIV(E, 256), 256, 0, stream>>>(ei[l], deg, E);
    k_deg_inv<<<(unsigned)CDIV(N, 256), 256, 0, stream>>>(deg, N);
    k_edge_norm<<<(unsigned)CDIV(E, 256), 256, 0, stream>>>(ei[l], deg, nrm[l], E);
  }

  // ---- encoder: conv -> relu -> pool, 4 levels ----
  const float* cur = x;
  for (int i = 0; i < 4; ++i) {
    run_conv(cur, ei[i], nrm[i], encW[i], encB[i], actA, tb0, tb1, tb2,
             NSs[i], FE[i], FE[i + 1], /*relu=*/1, stream);
    long long pt = 4LL * NSs[i + 1] * FE[i + 1];
    k_pool<<<(unsigned)CDIV(pt, 256), 256, 0, stream>>>(
        actA, down_i[i], down_w[i], actB, NSs[i + 1], NSs[i], FE[i + 1]);
    cur = actB;
  }

  // ---- dense bottleneck ----
  k_dense_enc<<<256, 256, 0, stream>>>(actB, enc_w, enc_b, zbuf);
  k_dense_dec<<<(unsigned)CDIV(4LL * 8192, 256), 256, 0, stream>>>(zbuf, dec_w,
                                                                   dec_b, actA);

  // ---- decoder: pool up -> conv -> relu, 4 levels ----
  for (int i = 0; i < 4; ++i) {
    int lvl = 3 - i;
    long long pt = 4LL * NSs[lvl] * FD[i];
    k_pool<<<(unsigned)CDIV(pt, 256), 256, 0, stream>>>(
        actA, up_i[lvl], up_w[lvl], actB, NSs[lvl], NSs[lvl + 1], FD[i]);
    run_conv(actB, ei[lvl], nrm[lvl], decW[i], decB[i], actA, tb0, tb1, tb2,
             NSs[lvl], FD[i], FD[i + 1], /*relu=*/1, stream);
  }

  // ---- final conv (16 -> 3, no bias, no relu) straight into d_out ----
  run_conv(actA, ei[0], nrm[0], decW[4], nullptr, (float*)d_out, tb0, tb1, tb2,
           NSs[0], FD[4], FD[5], /*relu=*/0, stream);
}